// NTXentLossLIDAug_33767032881855
// MI455X (gfx1250) — compile-verified
//
#include <hip/hip_runtime.h>
#include <math.h>

typedef __attribute__((ext_vector_type(2))) float v2f;
typedef __attribute__((ext_vector_type(8))) float v8f;
typedef int v4i_t __attribute__((vector_size(16)));               // int __vector(4)
typedef __attribute__((address_space(1))) v4i_t gv4i;             // global (AS1)
typedef __attribute__((address_space(3))) v4i_t lv4i;             // LDS (AS3)

#define BHALF 4096
#define NROW  8192        // 2*B
#define DZ    128
#define DF    2048
#define LCOLS 8191        // 2B-1
#define TEMP_INV 2.0f     // 1 / 0.5
#define NORM_EPS 1e-12f
#define LID_EPS  1e-12f
#define APAD  20          // LDS row stride in floats: 80B = 16B-aligned, conflict-free reads

#if __has_builtin(__builtin_amdgcn_global_load_async_to_lds_b128) && \
    __has_builtin(__builtin_amdgcn_s_wait_asynccnt)
#define HAVE_ASYNC_LDS 1
#else
#define HAVE_ASYNC_LDS 0
#endif

// ---------------------------------------------------------------- WMMA helper
__device__ __forceinline__ v8f wmma4(v2f a, v2f b, v8f c) {
  // V_WMMA_F32_16X16X4_F32: D = A(16x4,f32) * B(4x16,f32) + C(16x16,f32)
  return __builtin_amdgcn_wmma_f32_16x16x4_f32(
      /*neg_a=*/false, a, /*neg_b=*/false, b,
      /*c_mod=*/(short)0, c, /*reuse_a=*/false, /*reuse_b=*/false);
}

__device__ __forceinline__ const float* f_row(const float* f0, const float* f1, int g) {
  return (g < BHALF) ? (f0 + (size_t)g * DF) : (f1 + (size_t)(g - BHALF) * DF);
}

// ------------------------------------------------- K1: fsq + normalized Z rows
__global__ __launch_bounds__(256) void prep_kernel(const float* __restrict__ z0,
                                                   const float* __restrict__ z1,
                                                   const float* __restrict__ f0,
                                                   const float* __restrict__ f1,
                                                   float* __restrict__ fsq,
                                                   float* __restrict__ Z) {
  __shared__ float red[256];
  const int r = blockIdx.x;
  const int tid = threadIdx.x;

  const float* frow = f_row(f0, f1, r);
  float s = 0.f;
  for (int c = tid; c < DF; c += 256) { float v = frow[c]; s += v * v; }
  red[tid] = s;
  for (int off = 128; off > 0; off >>= 1) { __syncthreads(); if (tid < off) red[tid] += red[tid + off]; }
  __syncthreads();
  if (tid == 0) fsq[r] = red[0];
  __syncthreads();

  const float* zrow = (r < BHALF) ? (z0 + (size_t)r * DZ) : (z1 + (size_t)(r - BHALF) * DZ);
  float v = (tid < DZ) ? zrow[tid] : 0.f;
  red[tid] = v * v;
  for (int off = 128; off > 0; off >>= 1) { __syncthreads(); if (tid < off) red[tid] += red[tid + off]; }
  __syncthreads();
  float nrm = fmaxf(sqrtf(red[0]), NORM_EPS);
  if (tid < DZ) Z[(size_t)r * DZ + tid] = v / nrm;
}

// ----------------- K2: distance GEMM (f32 WMMA, 256x128 tile, async ping-pong)
__global__ __launch_bounds__(256) void dist_kernel(const float* __restrict__ f0,
                                                   const float* __restrict__ f1,
                                                   const float* __restrict__ fsq,
                                                   float* __restrict__ D) {
  __shared__ alignas(16) float At[2][256][APAD];   // 256 rows of A (i side)
  __shared__ alignas(16) float Bt[2][128][APAD];   // 128 rows of B (j side)
  const int rowbase = blockIdx.y * 256;
  const int colbase = blockIdx.x * 128;
  const int tid  = threadIdx.x;
  const int lane = tid & 31;
  const int wave = tid >> 5;
  const int lm   = lane & 15;
  const int lh   = lane >> 4;      // 0: K 0..1 / rows M, 1: K 2..3 / rows M+8
  v8f acc0[8] = {};
  v8f acc1[8] = {};

  // issue one 16-wide K slab into LDS buffer `buf` (6 x 16B chunks per thread)
  auto issue_tile = [&](int buf, int kk) {
    for (int q = tid; q < 1536; q += 256) {
      if (q < 1024) {                       // A: 256 rows x 4 chunks
        const int row = q >> 2;
        const int ko  = (q & 3) << 2;
        const float* src = f_row(f0, f1, rowbase + row) + kk + ko;
#if HAVE_ASYNC_LDS
        __builtin_amdgcn_global_load_async_to_lds_b128(
            (gv4i*)src, (lv4i*)&At[buf][row][ko], 0, 0);
#else
        const float4 va = *(const float4*)src;
        At[buf][row][ko + 0] = va.x; At[buf][row][ko + 1] = va.y;
        At[buf][row][ko + 2] = va.z; At[buf][row][ko + 3] = va.w;
#endif
      } else {                              // B: 128 rows x 4 chunks
        const int q2  = q - 1024;
        const int row = q2 >> 2;
        const int ko  = (q2 & 3) << 2;
        const float* src = f_row(f0, f1, colbase + row) + kk + ko;
#if HAVE_ASYNC_LDS
        __builtin_amdgcn_global_load_async_to_lds_b128(
            (gv4i*)src, (lv4i*)&Bt[buf][row][ko], 0, 0);
#else
        const float4 vb = *(const float4*)src;
        Bt[buf][row][ko + 0] = vb.x; Bt[buf][row][ko + 1] = vb.y;
        Bt[buf][row][ko + 2] = vb.z; Bt[buf][row][ko + 3] = vb.w;
#endif
      }
    }
  };

  int buf = 0;
  issue_tile(0, 0);
  for (int kk = 0; kk < DF; kk += 16) {
#if HAVE_ASYNC_LDS
    if (kk + 16 < DF) {
      issue_tile(buf ^ 1, kk + 16);                 // prefetch next slab
      __builtin_amdgcn_s_wait_asynccnt(6);          // current slab's 6 loads done
    } else {
      __builtin_amdgcn_s_wait_asynccnt(0);
    }
#endif
    __syncthreads();                                // all waves' slab visible

    const int m0 = wave * 32;
#pragma unroll
    for (int k4 = 0; k4 < 4; ++k4) {
      const int kb = k4 * 4 + lh * 2;
      v2f a0, a1;
      a0.x = At[buf][m0      + lm][kb]; a0.y = At[buf][m0      + lm][kb + 1];
      a1.x = At[buf][m0 + 16 + lm][kb]; a1.y = At[buf][m0 + 16 + lm][kb + 1];
#pragma unroll
      for (int j = 0; j < 8; ++j) {
        v2f b; b.x = Bt[buf][j * 16 + lm][kb]; b.y = Bt[buf][j * 16 + lm][kb + 1];
        acc0[j] = wmma4(a0, b, acc0[j]);
        acc1[j] = wmma4(a1, b, acc1[j]);
      }
    }
    __syncthreads();                                // done reading before overwrite
#if HAVE_ASYNC_LDS
    buf ^= 1;
#else
    if (kk + 16 < DF) issue_tile(buf, kk + 16);     // single-buffer fallback
#endif
  }

#pragma unroll
  for (int j = 0; j < 8; ++j) {
    const int gj = colbase + j * 16 + lm;
    const float fsj = fsq[gj];
#pragma unroll
    for (int d = 0; d < 8; ++d) {
      const int gi0 = rowbase + wave * 32 + d + lh * 8;
      const int gi1 = gi0 + 16;
      const float d20 = fsq[gi0] + fsj - 2.f * acc0[j][d];
      const float d21 = fsq[gi1] + fsj - 2.f * acc1[j][d];
      D[(size_t)gi0 * NROW + gj] = sqrtf(fmaxf(d20, 0.f));
      D[(size_t)gi1 * NROW + gj] = sqrtf(fmaxf(d21, 0.f));
    }
  }
}

// ------------------- K3: per-row 513-NN selection (bitonic sort in LDS) + LID
__global__ __launch_bounds__(256) void lid_kernel(const float* __restrict__ D,
                                                  float* __restrict__ lid32,
                                                  float* __restrict__ lid512) {
  __shared__ float sd[NROW];
  __shared__ float red[256];
  const int r = blockIdx.x;
  const int tid = threadIdx.x;
  const float* drow = D + (size_t)r * NROW;
  for (int i = tid; i < NROW; i += 256) sd[i] = drow[i];
  __syncthreads();

  for (int k = 2; k <= NROW; k <<= 1) {
    for (int j = k >> 1; j > 0; j >>= 1) {
      for (int i = tid; i < NROW; i += 256) {
        const int ixj = i ^ j;
        if (ixj > i) {
          const float a = sd[i], b = sd[ixj];
          const bool up = ((i & k) == 0);
          if ((a > b) == up) { sd[i] = b; sd[ixj] = a; }
        }
      }
      __syncthreads();
    }
  }
  // sd ascending; sd[0] is the zero self-distance. d_j = sd[1..k], d_k = sd[k].
  const float dk512 = sd[512];
  float l = 0.f;
  for (int j = 1 + tid; j <= 512; j += 256) l += logf(sd[j] / dk512 + LID_EPS);
  red[tid] = l;
  for (int off = 128; off > 0; off >>= 1) { __syncthreads(); if (tid < off) red[tid] += red[tid + off]; }
  __syncthreads();
  const float s512 = red[0];
  __syncthreads();
  const float dk32 = sd[32];
  red[tid] = (tid < 32) ? logf(sd[1 + tid] / dk32 + LID_EPS) : 0.f;
  for (int off = 128; off > 0; off >>= 1) { __syncthreads(); if (tid < off) red[tid] += red[tid + off]; }
  __syncthreads();
  if (tid == 0) {
    lid32[r]  = -32.f  / red[0];
    lid512[r] = -512.f / s512;
  }
}

// ------------- K4: logits GEMM (Z Z^T / T, f32 WMMA) with off-diag scatter
__global__ __launch_bounds__(256) void logits_kernel(const float* __restrict__ Z,
                                                     float* __restrict__ out) {
  __shared__ float At[128][17];
  __shared__ float Bt[128][17];
  const int rowbase = blockIdx.y * 128;
  const int colbase = blockIdx.x * 128;
  const int tid = threadIdx.x;
  const int lane = tid & 31;
  const int wave = tid >> 5;
  const int lm = lane & 15;
  const int lh = lane >> 4;
  v8f acc[8] = {};

  for (int kk = 0; kk < DZ; kk += 16) {
    for (int q = tid; q < 512; q += 256) {
      const int row = q >> 2;
      const int ko  = (q & 3) << 2;
      const float4 va = *(const float4*)(Z + (size_t)(rowbase + row) * DZ + kk + ko);
      At[row][ko + 0] = va.x; At[row][ko + 1] = va.y;
      At[row][ko + 2] = va.z; At[row][ko + 3] = va.w;
      const float4 vb = *(const float4*)(Z + (size_t)(colbase + row) * DZ + kk + ko);
      Bt[row][ko + 0] = vb.x; Bt[row][ko + 1] = vb.y;
      Bt[row][ko + 2] = vb.z; Bt[row][ko + 3] = vb.w;
    }
    __syncthreads();
    const int m0 = wave * 16;
#pragma unroll
    for (int k4 = 0; k4 < 4; ++k4) {
      const int kb = k4 * 4 + lh * 2;
      v2f a; a.x = At[m0 + lm][kb]; a.y = At[m0 + lm][kb + 1];
#pragma unroll
      for (int j = 0; j < 8; ++j) {
        v2f b; b.x = Bt[j * 16 + lm][kb]; b.y = Bt[j * 16 + lm][kb + 1];
        acc[j] = wmma4(a, b, acc[j]);
      }
    }
    __syncthreads();
  }

#pragma unroll
  for (int j = 0; j < 8; ++j) {
    const int c = colbase + j * 16 + lm;
#pragma unroll
    for (int d = 0; d < 8; ++d) {
      const int r = rowbase + wave * 16 + d + lh * 8;
      float v = TEMP_INV * acc[j][d];
      if (!__builtin_isfinite(v)) v = 0.f;   // nan_to_num
      int dest; bool skip = false;
      if (r < BHALF) {
        if (c >= BHALF)      dest = c - BHALF;                         // logits_01
        else if (c == r)     skip = true;                              // diagonal of 00
        else                 dest = BHALF + (c < r ? c : c - 1);       // offdiag 00
      } else {
        const int i = r - BHALF;
        if (c < BHALF)       dest = c;                                 // logits_10
        else {
          const int cc = c - BHALF;
          if (cc == i)       skip = true;                              // diagonal of 11
          else               dest = BHALF + (cc < i ? cc : cc - 1);    // offdiag 11
        }
      }
      if (!skip) out[1 + (size_t)r * LCOLS + dest] = v;
    }
  }
}

// ------------------------------------------------------------- K5: labels
__global__ void labels_kernel(float* __restrict__ out) {
  const int i = blockIdx.x * blockDim.x + threadIdx.x;
  if (i < NROW) out[1 + (size_t)NROW * LCOLS + i] = (float)(i & (BHALF - 1));
}

// ----------------------------- K6: per-row online log-softmax -> loss_per[r]
__global__ __launch_bounds__(256) void loss_row_kernel(const float* __restrict__ logits,
                                                       const float* __restrict__ Z,
                                                       float* __restrict__ loss_per) {
  __shared__ float sm[256];
  __shared__ float ss[256];
  const int r = blockIdx.x;
  const int tid = threadIdx.x;
  const float* row = logits + (size_t)r * LCOLS;
  float m = -3.402823e38f, s = 0.f;
  for (int c = tid; c < LCOLS; c += 256) {
    const float x = row[c];
    if (x > m) { s = s * expf(m - x) + 1.f; m = x; } else { s += expf(x - m); }
  }
  sm[tid] = m; ss[tid] = s;
  for (int off = 128; off > 0; off >>= 1) {
    __syncthreads();
    if (tid < off) {
      const float m1 = sm[tid], m2 = sm[tid + off];
      const float s1 = ss[tid], s2 = ss[tid + off];
      const float mm = fmaxf(m1, m2);
      sm[tid] = mm;
      ss[tid] = s1 * expf(m1 - mm) + s2 * expf(m2 - mm);
    }
  }
  __syncthreads();
  const float lse = sm[0] + logf(ss[0]);
  __syncthreads();
  const int i = r & (BHALF - 1);
  float p = 0.f;
  if (tid < DZ) p = Z[(size_t)i * DZ + tid] * Z[(size_t)(BHALF + i) * DZ + tid];
  ss[tid] = p;
  for (int off = 128; off > 0; off >>= 1) { __syncthreads(); if (tid < off) ss[tid] += ss[tid + off]; }
  __syncthreads();
  if (tid == 0) {
    const float pos = TEMP_INV * ss[0];
    float lp = lse - pos;               // -(pos - lse)
    if (!__builtin_isfinite(lp)) lp = 0.1f;
    loss_per[r] = lp;
  }
}

// ------------------------------------------------------- K7: mean of losses
__global__ __launch_bounds__(256) void final_kernel(const float* __restrict__ loss_per,
                                                    float* __restrict__ out) {
  __shared__ float red[256];
  const int tid = threadIdx.x;
  float s = 0.f;
  for (int i = tid; i < NROW; i += 256) s += loss_per[i];
  red[tid] = s;
  for (int off = 128; off > 0; off >>= 1) { __syncthreads(); if (tid < off) red[tid] += red[tid + off]; }
  __syncthreads();
  if (tid == 0) out[0] = red[0] / (float)NROW;
}

// ---------------------------------------------------------------------------
extern "C" void kernel_launch(void* const* d_in, const int* in_sizes, int n_in,
                              void* d_out, int out_size, void* d_ws, size_t ws_size,
                              hipStream_t stream) {
  (void)in_sizes; (void)n_in; (void)out_size; (void)ws_size;
  const float* z0 = (const float*)d_in[0];
  const float* z1 = (const float*)d_in[1];
  const float* f0 = (const float*)d_in[2];
  const float* f1 = (const float*)d_in[3];
  float* out = (float*)d_out;

  float* w        = (float*)d_ws;
  float* fsq      = w;                              // 8192
  float* Z        = w + NROW;                       // 8192*128
  float* loss_per = Z + (size_t)NROW * DZ;          // 8192   (~4.3 MB total)

  // D scratch aliases the logits+labels region of d_out (exactly 8192^2 floats)
  float* D      = out + 1;
  float* lid32  = out + 1 + (size_t)NROW * LCOLS + NROW;
  float* lid512 = lid32 + NROW;

  prep_kernel  <<<NROW, 256, 0, stream>>>(z0, z1, f0, f1, fsq, Z);
  dist_kernel  <<<dim3(64, 32), 256, 0, stream>>>(f0, f1, fsq, D);
  lid_kernel   <<<NROW, 256, 0, stream>>>(D, lid32, lid512);
  logits_kernel<<<dim3(64, 64), 256, 0, stream>>>(Z, out);          // overwrites D region
  labels_kernel<<<(NROW + 255) / 256, 256, 0, stream>>>(out);
  loss_row_kernel<<<NROW, 256, 0, stream>>>(out + 1, Z, loss_per);
  final_kernel <<<1, 256, 0, stream>>>(loss_per, out);
}